// CircleLossSoftplus_41240275976792
// MI455X (gfx1250) — compile-verified
//
#include <hip/hip_runtime.h>
#include <hip/hip_bf16.h>
#include <math.h>

// ---------------------------------------------------------------------------
// Circle loss (softplus(logsumexp_n + logsumexp_p)) fused with f32 WMMA GEMM.
// N=8192, D=128, 100 classes, m=0.25, gamma=256.
// Round 2: triangular-only dispatch + 32x32 register-blocked wave tiles
// (2x2 sub-tiles -> 4 independent WMMA accumulation chains, 2x less L2
// operand traffic, 2x better wmma:load ratio).
// ---------------------------------------------------------------------------

typedef float v2f __attribute__((ext_vector_type(2)));
typedef float v8f __attribute__((ext_vector_type(8)));

#define N_ROWS 8192
#define DIM    128
#define NT32   (N_ROWS / 32)                 // 256 32-wide tiles per dimension
#define NTILES (NT32 * (NT32 + 1) / 2)       // 32896 upper-triangle wave tiles
#define WPB    8                             // waves per block
#define NBLK2  (NTILES / WPB)                // 4112 blocks (exact: 32896 = 8*4112)

#define M_MARGIN 0.25f
#define GAMMA    256.0f
#define NEG_INF  (-__builtin_inff())

// Online logsumexp: push one value x into running (m, s) where lse = m+log(s).
// Safe for m == -inf (exp(-inf) == 0 -> s becomes 1).
__device__ __forceinline__ void lse_push(float& m, float& s, float x) {
    if (x > m) {
        s = s * __expf(m - x) + 1.0f;
        m = x;
    } else {
        s = s + __expf(x - m);
    }
}

// Merge (m2, s2) into (m, s). Branches guard the -inf - -inf = NaN case.
__device__ __forceinline__ void lse_merge(float& m, float& s, float m2, float s2) {
    if (m2 == NEG_INF) return;
    if (m == NEG_INF) { m = m2; s = s2; return; }
    if (m2 > m) {
        s = s * __expf(m - m2) + s2;
        m = m2;
    } else {
        s = s + s2 * __expf(m2 - m);
    }
}

// Circle-loss logit for one sim element, pushed into the right accumulator.
__device__ __forceinline__ void circle_push(float s, int row, int col, int li, int lj,
                                            float& mP, float& sP, float& mN, float& sN) {
    if (row < col) {                           // strict upper triangle
        if (li == lj) {
            // ap = max(1 + m - s, 0); logit_p = -ap*(s - (1-m))*gamma
            const float ap = fmaxf((1.0f + M_MARGIN) - s, 0.0f);
            lse_push(mP, sP, -GAMMA * ap * (s - (1.0f - M_MARGIN)));
        } else {
            // an = max(s + m, 0); logit_n = an*(s - m)*gamma
            const float an = fmaxf(s + M_MARGIN, 0.0f);
            lse_push(mN, sN, GAMMA * an * (s - M_MARGIN));
        }
    }
}

// Largest b with f(b) = b*n - b*(b-1)/2 <= t (upper-triangle row decode).
__device__ __forceinline__ int tri_row(int t, int n) {
    const double nn = 2.0 * (double)n + 1.0;
    int b = (int)((nn - sqrt(nn * nn - 8.0 * (double)t)) * 0.5);
    if (b < 0) b = 0;
    if (b > n - 1) b = n - 1;
    while ((long)(b + 1) * n - ((long)(b + 1) * b) / 2 <= (long)t) ++b;
    while ((long)b * n - ((long)b * (b - 1)) / 2 > (long)t) --b;
    return b;
}

// ---------------------------------------------------------------------------
// Kernel 1: L2-normalize each row. One wave (32 lanes) per 128-float row.
// ---------------------------------------------------------------------------
__global__ void __launch_bounds__(256)
normalize_kernel(const float* __restrict__ in, float* __restrict__ out) {
    const int row  = (blockIdx.x * blockDim.x + threadIdx.x) >> 5;
    const int lane = threadIdx.x & 31;
    if (row >= N_ROWS) return;

    const float4* src = (const float4*)(in + (size_t)row * DIM);
    float4 v = src[lane];
    float ss = v.x * v.x + v.y * v.y + v.z * v.z + v.w * v.w;
    #pragma unroll
    for (int off = 16; off; off >>= 1)
        ss += __shfl_xor(ss, off, 32);

    const float scale = 1.0f / fmaxf(sqrtf(ss), 1e-12f);
    float4 o;
    o.x = v.x * scale; o.y = v.y * scale; o.z = v.z * scale; o.w = v.w * scale;
    ((float4*)(out + (size_t)row * DIM))[lane] = o;
}

// ---------------------------------------------------------------------------
// Kernel 2: one 32x32 sim tile per wave as 2x2 16x16 WMMA sub-tiles.
// Per K-step (K=4): 4 float2 operand loads feed 4 V_WMMA_F32_16X16X4_F32
// (independent accumulation chains). Fused circle-loss epilogue with online
// logsumexp; block partials written as float4.
//
// f32 16x4 A layout: lanes 0-15 hold M=0..15 with VGPR0=K0, VGPR1=K1;
// lanes 16-31 hold K2/K3 -> contiguous float2 at F[row] + kk + 2*half.
// C/D: acc[r] = sim[rowBase(+16) + r + 8*half][colBase(+16) + (lane&15)].
// ---------------------------------------------------------------------------
__global__ void __launch_bounds__(256)
circle_tile_kernel(const float* __restrict__ F,
                   const int* __restrict__ labels,
                   float4* __restrict__ partials) {
    const int wave = threadIdx.x >> 5;
    const int lane = threadIdx.x & 31;

    // Linear index over upper-triangle (bi <= bj) 32x32 tiles: no dead waves.
    const int t  = blockIdx.x * WPB + wave;          // t < NTILES by construction
    const int bi = tri_row(t, NT32);
    const long fb = (long)bi * NT32 - ((long)bi * (bi - 1)) / 2;
    const int bj = bi + (int)(t - fb);

    const int half = lane >> 4;
    const int l15  = lane & 15;
    const int rowBase = bi * 32;
    const int colBase = bj * 32;

    const v2f* A0 = (const v2f*)(F + (size_t)(rowBase + l15)      * DIM);
    const v2f* A1 = (const v2f*)(F + (size_t)(rowBase + 16 + l15) * DIM);
    const v2f* B0 = (const v2f*)(F + (size_t)(colBase + l15)      * DIM);
    const v2f* B1 = (const v2f*)(F + (size_t)(colBase + 16 + l15) * DIM);

    v8f acc00 = {}, acc01 = {}, acc10 = {}, acc11 = {};
    #pragma unroll
    for (int kk = 0; kk < DIM; kk += 4) {
        const int idx = (kk >> 1) + half;
        const v2f a0 = A0[idx];
        const v2f a1 = A1[idx];
        const v2f b0 = B0[idx];
        const v2f b1 = B1[idx];
        acc00 = __builtin_amdgcn_wmma_f32_16x16x4_f32(false, a0, false, b0, (short)0, acc00, false, false);
        acc01 = __builtin_amdgcn_wmma_f32_16x16x4_f32(false, a0, false, b1, (short)0, acc01, false, false);
        acc10 = __builtin_amdgcn_wmma_f32_16x16x4_f32(false, a1, false, b0, (short)0, acc10, false, false);
        acc11 = __builtin_amdgcn_wmma_f32_16x16x4_f32(false, a1, false, b1, (short)0, acc11, false, false);
    }

    // Fused epilogue: online logsumexp over the 4x8 = 32 elements of this lane.
    float mP = NEG_INF, sP = 0.0f;
    float mN = NEG_INF, sN = 0.0f;

    const int col0 = colBase + l15;
    const int col1 = colBase + 16 + l15;
    const int lj0 = labels[col0];
    const int lj1 = labels[col1];

    #pragma unroll
    for (int r = 0; r < 8; ++r) {
        const int rowA = rowBase + r + 8 * half;
        const int rowB = rowA + 16;
        const int liA = labels[rowA];
        const int liB = labels[rowB];
        circle_push(acc00[r], rowA, col0, liA, lj0, mP, sP, mN, sN);
        circle_push(acc01[r], rowA, col1, liA, lj1, mP, sP, mN, sN);
        circle_push(acc10[r], rowB, col0, liB, lj0, mP, sP, mN, sN);
        circle_push(acc11[r], rowB, col1, liB, lj1, mP, sP, mN, sN);
    }

    // Wave32 shuffle reduction of the two (m, s) accumulators.
    #pragma unroll
    for (int off = 16; off; off >>= 1) {
        float m2 = __shfl_xor(mP, off, 32);
        float s2 = __shfl_xor(sP, off, 32);
        lse_merge(mP, sP, m2, s2);
        m2 = __shfl_xor(mN, off, 32);
        s2 = __shfl_xor(sN, off, 32);
        lse_merge(mN, sN, m2, s2);
    }

    __shared__ float4 smem[WPB];
    if (lane == 0) smem[wave] = make_float4(mP, sP, mN, sN);
    __syncthreads();

    if (threadIdx.x == 0) {
        float MP = NEG_INF, SP = 0.0f, MN = NEG_INF, SN = 0.0f;
        #pragma unroll
        for (int w = 0; w < WPB; ++w) {
            const float4 p = smem[w];
            lse_merge(MP, SP, p.x, p.y);
            lse_merge(MN, SN, p.z, p.w);
        }
        partials[blockIdx.x] = make_float4(MP, SP, MN, SN);
    }
}

// ---------------------------------------------------------------------------
// Kernel 3: merge all block partials, softplus, write scalar.
// ---------------------------------------------------------------------------
__global__ void __launch_bounds__(256)
final_reduce_kernel(const float4* __restrict__ partials, float* __restrict__ out) {
    float mP = NEG_INF, sP = 0.0f;
    float mN = NEG_INF, sN = 0.0f;

    for (int i = threadIdx.x; i < NBLK2; i += 256) {
        const float4 p = partials[i];
        lse_merge(mP, sP, p.x, p.y);
        lse_merge(mN, sN, p.z, p.w);
    }

    const int lane = threadIdx.x & 31;
    const int wave = threadIdx.x >> 5;
    #pragma unroll
    for (int off = 16; off; off >>= 1) {
        float m2 = __shfl_xor(mP, off, 32);
        float s2 = __shfl_xor(sP, off, 32);
        lse_merge(mP, sP, m2, s2);
        m2 = __shfl_xor(mN, off, 32);
        s2 = __shfl_xor(sN, off, 32);
        lse_merge(mN, sN, m2, s2);
    }

    __shared__ float4 smem[8];
    if (lane == 0) smem[wave] = make_float4(mP, sP, mN, sN);
    __syncthreads();

    if (threadIdx.x == 0) {
        float MP = NEG_INF, SP = 0.0f, MN = NEG_INF, SN = 0.0f;
        #pragma unroll
        for (int w = 0; w < 8; ++w) {
            const float4 p = smem[w];
            lse_merge(MP, SP, p.x, p.y);
            lse_merge(MN, SN, p.z, p.w);
        }
        const float logP = MP + __logf(SP);
        const float logN = MN + __logf(SN);
        const float x = logN + logP;
        // softplus with overflow guard
        out[0] = (x > 20.0f) ? x : log1pf(__expf(x));
    }
}

// ---------------------------------------------------------------------------
extern "C" void kernel_launch(void* const* d_in, const int* in_sizes, int n_in,
                              void* d_out, int out_size, void* d_ws, size_t ws_size,
                              hipStream_t stream) {
    const float* feats  = (const float*)d_in[0];
    const int*   labels = (const int*)d_in[1];
    float*       out    = (float*)d_out;

    // Workspace layout: [0, 4MB) normalized features, then block partials.
    float*  Fn       = (float*)d_ws;
    float4* partials = (float4*)((char*)d_ws + (size_t)N_ROWS * DIM * sizeof(float));

    // 1) normalize: one wave per row -> 8192 waves -> 1024 blocks of 256.
    normalize_kernel<<<N_ROWS / 8, 256, 0, stream>>>(feats, Fn);

    // 2) fused WMMA GEMM + circle-loss logsumexp partials (triangle only).
    circle_tile_kernel<<<NBLK2, 256, 0, stream>>>(Fn, labels, partials);

    // 3) final scalar.
    final_reduce_kernel<<<1, 256, 0, stream>>>(partials, out);
}